// CrossSliceAttention_91096256348396
// MI455X (gfx1250) — compile-verified
//
#include <hip/hip_runtime.h>
#include <hip/hip_bf16.h>

// ---------------------------------------------------------------------------
// CDNA5 (gfx1250, wave32) banded self-attention block:
//   Q/K/V = x @ W^T + b   (three 16384x1024 x 1024x1024 GEMMs, f32 WMMA,
//                          async global->LDS double-buffered staging)
//   scores banded to |s-t|<=k, softmax, out = attn@V + x, LayerNorm
// Outputs: out (B,S,D) f32 then attn (B,S,S) f32, concatenated in d_out.
// ---------------------------------------------------------------------------

typedef __attribute__((ext_vector_type(2))) float v2f;
typedef __attribute__((ext_vector_type(8))) float v8f;
typedef __attribute__((ext_vector_type(4))) int   v4i;

typedef __attribute__((address_space(1))) v4i gv4i;   // global-space 128b chunk
typedef __attribute__((address_space(3))) v4i lv4i;   // LDS-space    128b chunk

#define BATCH  8
#define SEQ    2048
#define DMODEL 1024
#define MTOT   (BATCH * SEQ)        // 16384 rows
#define LDA    36                   // padded LDS row stride (floats)
#define KTILE  32
#define NKT    (DMODEL / KTILE)     // 32 K-tiles

// ---- async global -> LDS (ASYNCcnt path) ----------------------------------
__device__ __forceinline__ void async_ld_b128(const void* gp, unsigned lds_off)
{
#if __has_builtin(__builtin_amdgcn_global_load_async_to_lds_b128)
    __builtin_amdgcn_global_load_async_to_lds_b128(
        (gv4i*)(uintptr_t)gp,
        (lv4i*)(uintptr_t)lds_off,
        0, 0);
#else
    asm volatile("global_load_async_to_lds_b128 %0, %1, off"
                 :: "v"(lds_off), "v"(gp) : "memory");
#endif
}

template <int N>
__device__ __forceinline__ void wait_asynccnt()
{
#if __has_builtin(__builtin_amdgcn_s_wait_asynccnt)
    __builtin_amdgcn_s_wait_asynccnt(N);
#else
    asm volatile("s_wait_asynccnt %0" :: "i"(N) : "memory");
#endif
}

// ---------------------------------------------------------------------------
// Kernel A: QKV projection via V_WMMA_F32_16X16X4_F32.
// Block = 128 threads (4 waves). Block tile: 64(M) x 64(N).
// Double-buffered LDS; tile t+1 streams in via async-to-LDS while tile t
// feeds the WMMA pipe. gridDim = (MTOT/64, DMODEL/64, 3).
// ---------------------------------------------------------------------------
__global__ void __launch_bounds__(128)
qkv_gemm_wmma(const float* __restrict__ x,
              const float* __restrict__ Wq, const float* __restrict__ bq,
              const float* __restrict__ Wk, const float* __restrict__ bk,
              const float* __restrict__ Wv, const float* __restrict__ bv,
              float* __restrict__ qkv_ws)
{
    __shared__ __align__(16) float As[2][64 * LDA];   // x tile [64 m][32 k]
    __shared__ __align__(16) float Bs[2][64 * LDA];   // W tile [64 n][32 k]

    const int z  = blockIdx.z;
    const float* __restrict__ W    = (z == 0) ? Wq : (z == 1) ? Wk : Wv;
    const float* __restrict__ bias = (z == 0) ? bq : (z == 1) ? bk : bv;
    float* __restrict__ out = qkv_ws + (size_t)z * (size_t)MTOT * DMODEL;

    const int m0 = blockIdx.x * 64;
    const int n0 = blockIdx.y * 64;

    const int t  = threadIdx.x;
    const int w  = t >> 5;            // wave id 0..3
    const int l  = t & 31;            // lane id
    const int lr = l & 15;            // row/col within lane half
    const int kh = (l < 16) ? 0 : 2;  // K offset for this half (A & B frags)

    // Per-thread staging coordinates: 4 float4 slots each for A and B tiles.
    int srow[4], sc4[4];
#pragma unroll
    for (int i = 0; i < 4; ++i) {
        const int f = i * 128 + t;    // 0..511 float4 slots
        srow[i] = f >> 3;             // 0..63
        sc4[i]  = (f & 7) << 2;       // 0,4,...,28
    }

    auto issue_tile = [&](int buf, int k0) {
#pragma unroll
        for (int i = 0; i < 4; ++i) {
            const float* ga = x + (size_t)(m0 + srow[i]) * DMODEL + k0 + sc4[i];
            const float* gb = W + (size_t)(n0 + srow[i]) * DMODEL + k0 + sc4[i];
            const int loff = srow[i] * LDA + sc4[i];
            async_ld_b128(ga, (unsigned)(uintptr_t)&As[buf][loff]);
            async_ld_b128(gb, (unsigned)(uintptr_t)&Bs[buf][loff]);
        }
    };

    // Accumulators with bias pre-folded (column depends only on n).
    v8f acc[4];
#pragma unroll
    for (int nt = 0; nt < 4; ++nt) {
        const float bv_ = bias[n0 + nt * 16 + lr];
#pragma unroll
        for (int j = 0; j < 8; ++j) acc[nt][j] = bv_;
    }

    const int arowoff = (w * 16 + lr) * LDA + kh;
    int browoff[4];
#pragma unroll
    for (int nt = 0; nt < 4; ++nt) browoff[nt] = (nt * 16 + lr) * LDA + kh;

    issue_tile(0, 0);                             // prologue: tile 0 in flight

    for (int kt = 0; kt < NKT; ++kt) {
        if (kt + 1 < NKT) {                       // stream next tile while
            issue_tile((kt + 1) & 1, (kt + 1) * KTILE);   // computing this one
            wait_asynccnt<8>();                   // tile kt's 8 ops complete
        } else {
            wait_asynccnt<0>();
        }
        __syncthreads();

        const float* as = &As[kt & 1][0];
        const float* bs = &Bs[kt & 1][0];

        // Software-pipelined fragment loads: fetch kk+4 before wmma'ing kk.
        v2f a = *(const v2f*)&as[arowoff];
        v2f b[4];
#pragma unroll
        for (int nt = 0; nt < 4; ++nt) b[nt] = *(const v2f*)&bs[browoff[nt]];

#pragma unroll
        for (int kk = 0; kk < KTILE; kk += 4) {
            v2f an = a;
            v2f bn[4] = { b[0], b[1], b[2], b[3] };
            if (kk + 4 < KTILE) {
                an = *(const v2f*)&as[arowoff + kk + 4];
#pragma unroll
                for (int nt = 0; nt < 4; ++nt)
                    bn[nt] = *(const v2f*)&bs[browoff[nt] + kk + 4];
            }
#pragma unroll
            for (int nt = 0; nt < 4; ++nt)
                acc[nt] = __builtin_amdgcn_wmma_f32_16x16x4_f32(
                    false, a, false, b[nt], (short)0, acc[nt], false, false);
            a = an;
#pragma unroll
            for (int nt = 0; nt < 4; ++nt) b[nt] = bn[nt];
        }
        __syncthreads();                          // buffer safe to overwrite
    }

    // Writeback: acc[nt][j] = C[m = w*16 + j + (l<16?0:8)][n = nt*16 + lr]
    const int mrow = m0 + w * 16 + ((l < 16) ? 0 : 8);
#pragma unroll
    for (int nt = 0; nt < 4; ++nt) {
        const int col = n0 + nt * 16 + lr;
#pragma unroll
        for (int j = 0; j < 8; ++j)
            out[(size_t)(mrow + j) * DMODEL + col] = acc[nt][j];
    }
}

// ---------------------------------------------------------------------------
// Block-wide (256-thread / 8-wave) sum reduction. wave32 shuffles + LDS.
// ---------------------------------------------------------------------------
__device__ inline float block_reduce_sum(float v, volatile float* red)
{
    const int tid = threadIdx.x;
#pragma unroll
    for (int o = 16; o > 0; o >>= 1) v += __shfl_xor(v, o, 32);
    __syncthreads();                       // protect red[] reuse
    if ((tid & 31) == 0) red[tid >> 5] = v;
    __syncthreads();
    float s = 0.f;
#pragma unroll
    for (int i = 0; i < 8; ++i) s += red[i];
    return s;
}

// ---------------------------------------------------------------------------
// Kernel B: per (b,s) row — banded scores, softmax, attn@V + residual,
// LayerNorm, plus full attn row write (zeros outside the band).
// grid = (SEQ, BATCH), block = 256 threads; each thread owns 4 channels.
// ---------------------------------------------------------------------------
__global__ void __launch_bounds__(256)
band_attn_ln(const float* __restrict__ x,
             const float* __restrict__ Q, const float* __restrict__ K,
             const float* __restrict__ V,
             const float* __restrict__ gamma, const float* __restrict__ beta,
             const int* __restrict__ kptr,
             float* __restrict__ outp, float* __restrict__ attn)
{
    __shared__ float red[8];
    __shared__ float sc[32];

    const int s   = blockIdx.x;
    const int b   = blockIdx.y;
    const int tid = threadIdx.x;

    int kband = kptr[0];
    if (kband < 0)  kband = 0;
    if (kband > 15) kband = 15;

    const int t0 = (s - kband > 0) ? (s - kband) : 0;
    const int t1 = (s + kband < SEQ - 1) ? (s + kband) : (SEQ - 1);
    const int nv = t1 - t0 + 1;                       // <= 31 valid neighbors

    const size_t rowb = ((size_t)b * SEQ + s) * DMODEL;
    const float4 q4 = *(const float4*)(Q + rowb + tid * 4);

    // Banded scores: dot(Q[s], K[t]) / sqrt(D)
    for (int j = 0; j < nv; ++j) {
        const float4 k4 = *(const float4*)(K + ((size_t)b * SEQ + t0 + j) * DMODEL + tid * 4);
        float p = q4.x * k4.x + q4.y * k4.y + q4.z * k4.z + q4.w * k4.w;
        p = block_reduce_sum(p, red);
        if (tid == 0) sc[j] = p * (1.0f / 32.0f);     // sqrt(1024) = 32
    }
    __syncthreads();

    // Softmax over the band (redundant scalar work per lane)
    float mx = -3.402823466e+38f;
    for (int j = 0; j < nv; ++j) mx = fmaxf(mx, sc[j]);
    float ssum = 0.f;
    for (int j = 0; j < nv; ++j) ssum += __expf(sc[j] - mx);
    const float inv_sum = 1.0f / ssum;

    // out = attn @ V + x (residual)
    float4 o4 = *(const float4*)(x + rowb + tid * 4);
    for (int j = 0; j < nv; ++j) {
        const float wgt = __expf(sc[j] - mx) * inv_sum;
        const float4 v4 = *(const float4*)(V + ((size_t)b * SEQ + t0 + j) * DMODEL + tid * 4);
        o4.x += wgt * v4.x; o4.y += wgt * v4.y;
        o4.z += wgt * v4.z; o4.w += wgt * v4.w;
    }

    // LayerNorm over D=1024
    const float psum = o4.x + o4.y + o4.z + o4.w;
    const float psq  = o4.x * o4.x + o4.y * o4.y + o4.z * o4.z + o4.w * o4.w;
    const float tsum = block_reduce_sum(psum, red);
    const float tsq  = block_reduce_sum(psq, red);
    const float mu   = tsum * (1.0f / (float)DMODEL);
    const float var  = tsq * (1.0f / (float)DMODEL) - mu * mu;
    const float rstd = rsqrtf(var + 1e-5f);

    const float4 g4  = *(const float4*)(gamma + tid * 4);
    const float4 be4 = *(const float4*)(beta + tid * 4);
    float4 y4;
    y4.x = (o4.x - mu) * rstd * g4.x + be4.x;
    y4.y = (o4.y - mu) * rstd * g4.y + be4.y;
    y4.z = (o4.z - mu) * rstd * g4.z + be4.z;
    y4.w = (o4.w - mu) * rstd * g4.w + be4.w;
    *(float4*)(outp + rowb + tid * 4) = y4;

    // attn row: zero all S entries, then write the band
    float* arow = attn + ((size_t)b * SEQ + s) * (size_t)SEQ;
    const float4 z4 = make_float4(0.f, 0.f, 0.f, 0.f);
    *(float4*)(arow + tid * 8 + 0) = z4;              // 256 threads * 8 = 2048
    *(float4*)(arow + tid * 8 + 4) = z4;
    __syncthreads();
    if (tid < nv) arow[t0 + tid] = __expf(sc[tid] - mx) * inv_sum;
}

// ---------------------------------------------------------------------------
extern "C" void kernel_launch(void* const* d_in, const int* in_sizes, int n_in,
                              void* d_out, int out_size, void* d_ws, size_t ws_size,
                              hipStream_t stream)
{
    const float* x     = (const float*)d_in[0];
    const float* Wq    = (const float*)d_in[1];
    const float* bq    = (const float*)d_in[2];
    const float* Wk    = (const float*)d_in[3];
    const float* bk    = (const float*)d_in[4];
    const float* Wv    = (const float*)d_in[5];
    const float* bv    = (const float*)d_in[6];
    const float* gamma = (const float*)d_in[7];
    const float* beta  = (const float*)d_in[8];
    const int*   kptr  = (const int*)d_in[9];

    float* qkv = (float*)d_ws;                         // 3 x 64 MB scratch
    float* Q   = qkv;
    float* K   = qkv + (size_t)MTOT * DMODEL;
    float* V   = qkv + 2 * (size_t)MTOT * DMODEL;

    float* outp = (float*)d_out;                       // (B,S,D)
    float* attn = outp + (size_t)MTOT * DMODEL;        // (B,S,S)

    dim3 gA(MTOT / 64, DMODEL / 64, 3);
    qkv_gemm_wmma<<<gA, 128, 0, stream>>>(x, Wq, bq, Wk, bk, Wv, bv, qkv);

    dim3 gB(SEQ, BATCH);
    band_attn_ln<<<gB, 256, 0, stream>>>(x, Q, K, V, gamma, beta, kptr, outp, attn);
}